// GNNActor_11845519803073
// MI455X (gfx1250) — compile-verified
//
#include <hip/hip_runtime.h>
#include <math.h>

#define N_NODES 79
#define RAW_FEAT 26
#define D_IN 32
#define D_OUT 64
#define HIDDEN 256
#define N_EDGES 3000
#define BATCH 256
#define M_TOTAL (BATCH * N_NODES)   /* 20224, divisible by 16 */
#define KP1 112                     /* 97 padded up to multiple of 4 (and 16) */
#define LDS_LD 264                  /* padded LDS row stride */

typedef float v2f __attribute__((ext_vector_type(2)));
typedef float v8f __attribute__((ext_vector_type(8)));

// ---------------------------------------------------------------- WMMA helpers
__device__ inline v8f wmma_f32(v2f a, v2f b, v8f c) {
  // D = A(16x4) * B(4x16) + C, fp32 tensor path on CDNA5
  return __builtin_amdgcn_wmma_f32_16x16x4_f32(false, a, false, b, (short)0, c,
                                               false, false);
}

// A 16x4 fragment: lanes 0-15 -> M=lane, K=k0+{0,1}; lanes 16-31 -> M=lane-16, K=k0+2+{0,1}
__device__ inline v2f load_frag_a(const float* A, int lda, int m0, int k0, int lane) {
  const int m = m0 + (lane & 15);
  const int k = k0 + ((lane >> 4) << 1);
  v2f r; r.x = A[(size_t)m * lda + k]; r.y = A[(size_t)m * lda + k + 1]; return r;
}
// B 4x16 fragment (transposed analog of A layout)
__device__ inline v2f load_frag_b(const float* B, int ldb, int k0, int n0, int lane) {
  const int n = n0 + (lane & 15);
  const int k = k0 + ((lane >> 4) << 1);
  v2f r; r.x = B[(size_t)k * ldb + n]; r.y = B[(size_t)(k + 1) * ldb + n]; return r;
}

__device__ inline float wave_allsum(float v) {
#pragma unroll
  for (int o = 16; o > 0; o >>= 1) v += __shfl_xor(v, o, 32);
  return v;
}

// ---------------------------------------------------------------- prep kernels
__global__ void x_kernel(const float* __restrict__ state, const float* __restrict__ pos,
                         float* __restrict__ Xbuf) {
  const int i = blockIdx.x * blockDim.x + threadIdx.x;
  if (i >= M_TOTAL * D_IN) return;
  const int r = i >> 5, c = i & 31;
  const int n = r % N_NODES;
  Xbuf[i] = (c < RAW_FEAT) ? state[(size_t)r * RAW_FEAT + c]
                           : pos[n * 6 + (c - RAW_FEAT)];
}

__global__ void tot_kernel(const float* __restrict__ state, float* __restrict__ tot) {
  const int b = threadIdx.x;  // one thread per batch, deterministic serial sum
  float s = 0.f;
  for (int n = 0; n < N_NODES; ++n)
    s += state[(size_t)(b * N_NODES + n) * RAW_FEAT + 1];
  tot[b] = s;
}

__global__ void wcat_kernel(const float* __restrict__ Wq, const float* __restrict__ Wk,
                            const float* __restrict__ Wv, const float* __restrict__ Ws,
                            const float* __restrict__ bq, const float* __restrict__ bk,
                            const float* __restrict__ bv, const float* __restrict__ bs,
                            float* __restrict__ Wcat, float* __restrict__ bcat) {
  const int i = blockIdx.x * blockDim.x + threadIdx.x;
  if (i < 32 * 256) {
    const int k = i >> 8, j = i & 255;
    const int g = j >> 6, jj = j & 63;
    const float* W = (g == 0) ? Wq : (g == 1) ? Wk : (g == 2) ? Wv : Ws;
    Wcat[k * 256 + j] = W[jj * D_IN + k];    // transpose: [K=32][N=256]
  }
  if (i < 256) {
    const int g = i >> 6, jj = i & 63;
    const float* bb = (g == 0) ? bq : (g == 1) ? bk : (g == 2) ? bv : bs;
    bcat[i] = bb[jj];
  }
}

__global__ void w1p_kernel(const float* __restrict__ W1, float* __restrict__ W1p) {
  const int i = blockIdx.x * blockDim.x + threadIdx.x;
  if (i >= KP1 * 256) return;
  const int k = i >> 8, h = i & 255;
  W1p[i] = (k < 97) ? W1[h * 97 + k] : 0.f;  // transpose + zero-pad K to 112
}

__global__ void w2p_kernel(const float* __restrict__ W2, float* __restrict__ W2p) {
  const int i = blockIdx.x * blockDim.x + threadIdx.x;
  if (i >= 256 * 256) return;
  const int k = i >> 8, h = i & 255;
  W2p[i] = W2[h * 256 + k];
}

// deterministic CSR over dst (single thread: 6000 trivial iterations)
__global__ void csr_kernel(const int* __restrict__ ei, int* __restrict__ rowp,
                           int* __restrict__ csre) {
  int cnt[N_NODES];
  for (int n = 0; n < N_NODES; ++n) cnt[n] = 0;
  for (int e = 0; e < N_EDGES; ++e) cnt[ei[N_EDGES + e]]++;
  int run = 0;
  for (int n = 0; n < N_NODES; ++n) { rowp[n] = run; run += cnt[n]; cnt[n] = rowp[n]; }
  rowp[N_NODES] = run;
  for (int e = 0; e < N_EDGES; ++e) { const int d = ei[N_EDGES + e]; csre[cnt[d]++] = e; }
}

// ----------------------------------------- fused Q|K|V|skip projection (WMMA)
__global__ void qkvs_kernel(const float* __restrict__ X, const float* __restrict__ Wcat,
                            const float* __restrict__ bcat, float* __restrict__ QKVS) {
  const int m0 = blockIdx.x * 16;
  const int wave = threadIdx.x >> 5, lane = threadIdx.x & 31;
#pragma unroll
  for (int t = 0; t < 2; ++t) {
    const int n0 = (wave * 2 + t) * 16;
    v8f acc = {};
#pragma unroll
    for (int k0 = 0; k0 < D_IN; k0 += 4) {
      v2f a = load_frag_a(X, D_IN, m0, k0, lane);
      v2f b = load_frag_b(Wcat, 256, k0, n0, lane);
      acc = wmma_f32(a, b, acc);
    }
    const int n = n0 + (lane & 15);
    const int mb = m0 + (lane >> 4) * 8;
    const float bias = bcat[n];
#pragma unroll
    for (int r = 0; r < 8; ++r)
      QKVS[(size_t)(mb + r) * 256 + n] = acc[r] + bias;
  }
}

// --------------------------------- TransformerConv: one wave per (batch, node)
__global__ void __launch_bounds__(32)
attn_kernel(const float* __restrict__ QKVS, const float* __restrict__ ew,
            const float* __restrict__ We, const int* __restrict__ ei,
            const int* __restrict__ rowp, const int* __restrict__ csre,
            const float* __restrict__ Xbuf, const float* __restrict__ tot,
            float* __restrict__ Hbuf) {
  __shared__ float s_alpha[N_EDGES];  // worst-case all edges on one node
  const int bid = blockIdx.x;
  const int b = bid / N_NODES;
  const int n = bid - b * N_NODES;
  const int lane = threadIdx.x;       // lane holds features {lane, lane+32}
  const int row = b * N_NODES + n;
  const float* qr = QKVS + (size_t)row * 256;
  const float q0 = qr[lane], q1 = qr[lane + 32];
  const float we0 = We[lane], we1 = We[lane + 32];
  const int e0 = rowp[n], e1 = rowp[n + 1];

  float amax = -3.402823466e+38f;
  for (int i = e0; i < e1; ++i) {              // pass 1: alphas + running max
    const int eid = csre[i];
    const int src = ei[eid];
    const float w = ew[eid];
    const float* kr = QKVS + (size_t)(b * N_NODES + src) * 256 + 64;
    float p = q0 * (kr[lane] + w * we0) + q1 * (kr[lane + 32] + w * we1);
    p = wave_allsum(p) * 0.125f;               // 1/sqrt(64)
    s_alpha[i - e0] = p;
    amax = fmaxf(amax, p);
  }
  float agg0 = 0.f, agg1 = 0.f, den = 0.f;
  for (int i = e0; i < e1; ++i) {              // pass 2: softmax-weighted agg
    const int eid = csre[i];
    const int src = ei[eid];
    const float w = ew[eid];
    const float* vr = QKVS + (size_t)(b * N_NODES + src) * 256 + 128;
    const float ex = expf(s_alpha[i - e0] - amax);
    den += ex;
    agg0 += ex * (vr[lane] + w * we0);
    agg1 += ex * (vr[lane + 32] + w * we1);
  }
  const float inv = (e1 > e0) ? 1.f / (den + 1e-16f) : 0.f;
  const float* sr = qr + 192;                  // skip projection slot
  float o0 = fmaxf(agg0 * inv + sr[lane], 0.f);
  float o1 = fmaxf(agg1 * inv + sr[lane + 32], 0.f);

  float* hr = Hbuf + (size_t)row * KP1;        // [out1 | tot | x | zero-pad]
  hr[lane] = o0;
  hr[lane + 32] = o1;
  hr[65 + lane] = Xbuf[(size_t)row * 32 + lane];
  if (lane == 0) hr[64] = tot[b];
  if (lane < 15) hr[97 + lane] = 0.f;
}

// --------------- fused MLP: WMMA GEMM -> LN -> leaky -> WMMA(A from LDS) -> LN
__global__ void mlp12_kernel(const float* __restrict__ H,
                             const float* __restrict__ W1p, const float* __restrict__ b1,
                             const float* __restrict__ g1,  const float* __restrict__ be1,
                             const float* __restrict__ W2p, const float* __restrict__ b2,
                             const float* __restrict__ g2,  const float* __restrict__ be2,
                             float* __restrict__ H2) {
  __shared__ float t1[16 * LDS_LD];
  __shared__ float t2[16 * LDS_LD];
  const int m0 = blockIdx.x * 16;
  const int wave = threadIdx.x >> 5, lane = threadIdx.x & 31;

  // ---- layer 1: [16 x 112] @ [112 x 256] -> t1
#pragma unroll
  for (int t = 0; t < 2; ++t) {
    const int n0 = (wave * 2 + t) * 16;
    v8f acc = {};
#pragma unroll
    for (int k0 = 0; k0 < KP1; k0 += 4) {
      v2f a = load_frag_a(H, KP1, m0, k0, lane);
      v2f b = load_frag_b(W1p, 256, k0, n0, lane);
      acc = wmma_f32(a, b, acc);
    }
    const int n = n0 + (lane & 15);
    const int mb = (lane >> 4) * 8;
    const float bias = b1[n];
#pragma unroll
    for (int r = 0; r < 8; ++r) t1[(mb + r) * LDS_LD + n] = acc[r] + bias;
  }
  __syncthreads();
  // LayerNorm + leaky_relu in place (each wave owns rows 2w, 2w+1)
#pragma unroll
  for (int rr = 0; rr < 2; ++rr) {
    const int rowl = wave * 2 + rr;
    float vals[8], s = 0.f, s2 = 0.f;
#pragma unroll
    for (int i = 0; i < 8; ++i) {
      const float v = t1[rowl * LDS_LD + i * 32 + lane];
      vals[i] = v; s += v; s2 += v * v;
    }
    s = wave_allsum(s); s2 = wave_allsum(s2);
    const float mean = s * (1.f / 256.f);
    const float inv = rsqrtf(s2 * (1.f / 256.f) - mean * mean + 1e-5f);
#pragma unroll
    for (int i = 0; i < 8; ++i) {
      const int j = i * 32 + lane;
      float v = (vals[i] - mean) * inv * g1[j] + be1[j];
      t1[rowl * LDS_LD + j] = (v > 0.f) ? v : 0.01f * v;
    }
  }
  __syncthreads();

  // ---- layer 2: [16 x 256] @ [256 x 256], A fragments straight from LDS
#pragma unroll
  for (int t = 0; t < 2; ++t) {
    const int n0 = (wave * 2 + t) * 16;
    v8f acc = {};
    for (int k0 = 0; k0 < 256; k0 += 4) {
      if (k0 + 16 < 256) __builtin_prefetch(&W2p[(size_t)(k0 + 16) * 256], 0, 1);
      v2f a = load_frag_a(t1, LDS_LD, 0, k0, lane);
      v2f b = load_frag_b(W2p, 256, k0, n0, lane);
      acc = wmma_f32(a, b, acc);
    }
    const int n = n0 + (lane & 15);
    const int mb = (lane >> 4) * 8;
    const float bias = b2[n];
#pragma unroll
    for (int r = 0; r < 8; ++r) t2[(mb + r) * LDS_LD + n] = acc[r] + bias;
  }
  __syncthreads();
#pragma unroll
  for (int rr = 0; rr < 2; ++rr) {
    const int rowl = wave * 2 + rr;
    float vals[8], s = 0.f, s2 = 0.f;
#pragma unroll
    for (int i = 0; i < 8; ++i) {
      const float v = t2[rowl * LDS_LD + i * 32 + lane];
      vals[i] = v; s += v; s2 += v * v;
    }
    s = wave_allsum(s); s2 = wave_allsum(s2);
    const float mean = s * (1.f / 256.f);
    const float inv = rsqrtf(s2 * (1.f / 256.f) - mean * mean + 1e-5f);
#pragma unroll
    for (int i = 0; i < 8; ++i) {
      const int j = i * 32 + lane;
      float v = (vals[i] - mean) * inv * g2[j] + be2[j];
      H2[(size_t)(m0 + rowl) * 256 + j] = (v > 0.f) ? v : 0.01f * v;
    }
  }
}

// -------------------------------------------- head: softplus + per-batch norm
__global__ void head_kernel(const float* __restrict__ H2, const float* __restrict__ W3,
                            const float* __restrict__ b3, float* __restrict__ out) {
  const int b = blockIdx.x;
  const int t = threadIdx.x;
  __shared__ float sc[128];
  float conc = 0.f;
  if (t < N_NODES) {
    const float* row = H2 + (size_t)(b * N_NODES + t) * 256;
    float acc = b3[0];
    for (int k = 0; k < 256; ++k) acc = fmaf(row[k], W3[k], acc);
    conc = (acc > 0.f) ? acc + log1pf(expf(-acc)) : log1pf(expf(acc));  // softplus
  }
  sc[t] = (t < N_NODES) ? conc : 0.f;
  __syncthreads();
  for (int s = 64; s > 0; s >>= 1) { if (t < s) sc[t] += sc[t + s]; __syncthreads(); }
  const float den = sc[0] + 1e-20f;
  if (t < N_NODES) out[b * N_NODES + t] = conc / den;
}

// ---------------------------------------------------------------------- launch
extern "C" void kernel_launch(void* const* d_in, const int* in_sizes, int n_in,
                              void* d_out, int out_size, void* d_ws, size_t ws_size,
                              hipStream_t stream) {
  const float* state = (const float*)d_in[0];
  const float* pos   = (const float*)d_in[1];
  const float* ew    = (const float*)d_in[2];
  const float* Wq = (const float*)d_in[3];  const float* bq = (const float*)d_in[4];
  const float* Wk = (const float*)d_in[5];  const float* bk = (const float*)d_in[6];
  const float* Wv = (const float*)d_in[7];  const float* bv = (const float*)d_in[8];
  const float* We = (const float*)d_in[9];
  const float* Ws = (const float*)d_in[10]; const float* bs = (const float*)d_in[11];
  const float* W1 = (const float*)d_in[12]; const float* b1 = (const float*)d_in[13];
  const float* g1 = (const float*)d_in[14]; const float* be1 = (const float*)d_in[15];
  const float* W2 = (const float*)d_in[16]; const float* b2 = (const float*)d_in[17];
  const float* g2 = (const float*)d_in[18]; const float* be2 = (const float*)d_in[19];
  const float* W3 = (const float*)d_in[20]; const float* b3 = (const float*)d_in[21];
  const int*   ei = (const int*)d_in[22];
  float* out = (float*)d_out;

  float* p = (float*)d_ws;
  float* Xbuf = p;  p += (size_t)M_TOTAL * D_IN;    // 0.65 M
  float* QKVS = p;  p += (size_t)M_TOTAL * 256;     // 5.18 M
  float* Hbuf = p;  p += (size_t)M_TOTAL * KP1;     // 2.27 M
  float* H2   = p;  p += (size_t)M_TOTAL * 256;     // 5.18 M
  float* Wcat = p;  p += 32 * 256;
  float* bcat = p;  p += 256;
  float* W1p  = p;  p += KP1 * 256;
  float* W2p  = p;  p += 256 * 256;
  float* tot  = p;  p += 256;
  int* rowp = (int*)p;
  int* csre = rowp + (N_NODES + 1);

  // weight prep + x build + totals + CSR
  x_kernel<<<(M_TOTAL * D_IN + 255) / 256, 256, 0, stream>>>(state, pos, Xbuf);
  tot_kernel<<<1, BATCH, 0, stream>>>(state, tot);
  wcat_kernel<<<(32 * 256 + 255) / 256, 256, 0, stream>>>(Wq, Wk, Wv, Ws, bq, bk, bv, bs,
                                                          Wcat, bcat);
  w1p_kernel<<<(KP1 * 256 + 255) / 256, 256, 0, stream>>>(W1, W1p);
  w2p_kernel<<<(256 * 256 + 255) / 256, 256, 0, stream>>>(W2, W2p);
  csr_kernel<<<1, 1, 0, stream>>>(ei, rowp, csre);

  // fused Q|K|V|skip projection (WMMA, M=20224 K=32 N=256)
  qkvs_kernel<<<M_TOTAL / 16, 256, 0, stream>>>(Xbuf, Wcat, bcat, QKVS);

  // TransformerConv attention + ReLU + MLP-input assembly
  attn_kernel<<<M_TOTAL, 32, 0, stream>>>(QKVS, ew, We, ei, rowp, csre, Xbuf, tot, Hbuf);

  // fused MLP1+LN+leaky -> MLP2+LN+leaky (WMMA both layers, layer-2 A from LDS)
  mlp12_kernel<<<M_TOTAL / 16, 256, 0, stream>>>(Hbuf, W1p, b1, g1, be1,
                                                 W2p, b2, g2, be2, H2);

  // softplus head + per-batch normalization
  head_kernel<<<BATCH, 128, 0, stream>>>(H2, W3, b3, out);
}